// WindowTGNRegressor_67216238183277
// MI455X (gfx1250) — compile-verified
//
#include <hip/hip_runtime.h>
#include <hip/hip_bf16.h>

typedef __attribute__((ext_vector_type(2))) float v2f;
typedef __attribute__((ext_vector_type(8))) float v8f;

#define D_DIM 64
#define H_HEADS 2
#define C_CH 32
#define TD_DIM 16
#define ED_DIM 17

// ---- deterministic float-max via monotone uint key ----
__device__ __forceinline__ unsigned fkey(float f) {
    unsigned u = __float_as_uint(f);
    return (u & 0x80000000u) ? ~u : (u | 0x80000000u);
}
__device__ __forceinline__ float funkey(unsigned k) {
    return (k & 0x80000000u) ? __uint_as_float(k ^ 0x80000000u) : __uint_as_float(~k);
}
// ---- 32.32 fixed point for deterministic sums ----
__device__ __forceinline__ unsigned long long f2fix(float x) {
    return (unsigned long long)(long long)(x * 4294967296.0f);
}
__device__ __forceinline__ float fix2f(unsigned long long u) {
    return (float)((long long)u) * 2.3283064365386963e-10f;
}

// ---------------- K0: zero scratch region ----------------
__global__ void k_zero(unsigned long long* p, size_t n64) {
    size_t i = (size_t)blockIdx.x * blockDim.x + threadIdx.x;
    size_t stride = (size_t)gridDim.x * blockDim.x;
    for (; i < n64; i += stride) p[i] = 0ull;
}

// ---------------- K1: Q/K/V/Skip GEMMs via V_WMMA_F32_16X16X4_F32 ----------------
// block = 256 threads = 8 waves; wave (mat, rhalf) computes one of {Q,K,V,Skip}
// for a 16-row subtile of this block's 32-row tile.
// X tile is staged into LDS with the CDNA5 async global->LDS DMA path.
__global__ __launch_bounds__(256)
void k_qkvs(const float* __restrict__ X,
            const float* __restrict__ Wq, const float* __restrict__ bq,
            const float* __restrict__ Wk, const float* __restrict__ bk,
            const float* __restrict__ Wv, const float* __restrict__ bv,
            const float* __restrict__ Ws, const float* __restrict__ bs,
            float* __restrict__ Q, float* __restrict__ K,
            float* __restrict__ V, float* __restrict__ S, int N)
{
    // pitch 68 floats: rows stay 16B aligned for B128, A-frag reads conflict-free
    __shared__ __align__(16) float Xs[32][68];
    const int t  = threadIdx.x;
    const int m0 = blockIdx.x * 32;

    // stage 32x64 floats as 512 B128 transfers: 2 per thread, via async DMA
    for (int idx = t; idx < 32 * 16; idx += 256) {
        int r  = idx >> 4;
        int c4 = idx & 15;
        int gr = m0 + r;
        if (gr < N) {
            const float* gsrc = X + (size_t)gr * D_DIM + c4 * 4;
            unsigned lds = (unsigned)(uintptr_t)(&Xs[r][c4 * 4]);
            asm volatile("global_load_async_to_lds_b128 %0, %1, off"
                         :: "v"(lds), "v"(gsrc) : "memory");
        } else {
            float4 z = {0.f, 0.f, 0.f, 0.f};
            *(float4*)(&Xs[r][c4 * 4]) = z;
        }
    }
    asm volatile("s_wait_asynccnt 0x0" ::: "memory");
    __syncthreads();

    const int wave  = t >> 5;
    const int lane  = t & 31;
    const int mat   = wave & 3;     // which output matrix
    const int rhalf = wave >> 2;    // which 16-row subtile

    const float* W = Wq; const float* B = bq; float* O = Q;
    if (mat == 1)      { W = Wk; B = bk; O = K; }
    else if (mat == 2) { W = Wv; B = bv; O = V; }
    else if (mat == 3) { W = Ws; B = bs; O = S; }

    const int col  = lane & 15;          // A row within subtile / B,D column
    const int hi   = lane >> 4;          // half-wave: K pair 0,1 vs 2,3 ; D rows 0-7 vs 8-15
    const int srow = rhalf * 16 + col;   // LDS row for A fragment

    for (int n0 = 0; n0 < D_DIM; n0 += 16) {
        v8f acc = {0.f,0.f,0.f,0.f,0.f,0.f,0.f,0.f};
        for (int k = 0; k < D_DIM; k += 4) {
            int kk = k + hi * 2;
            v2f a; a.x = Xs[srow][kk];
                   a.y = Xs[srow][kk + 1];
            v2f b; b.x = W[(size_t)kk * D_DIM + n0 + col];
                   b.y = W[(size_t)(kk + 1) * D_DIM + n0 + col];
            acc = __builtin_amdgcn_wmma_f32_16x16x4_f32(
                false, a, false, b, (short)0, acc, false, false);
        }
        float bias = B[n0 + col];
        #pragma unroll
        for (int r = 0; r < 8; ++r) {
            int orow = m0 + rhalf * 16 + r + hi * 8;
            if (orow < N) O[(size_t)orow * D_DIM + n0 + col] = acc[r] + bias;
        }
    }
}

// ---------------- K2: G[node][h][j] = sum_c We[j][h*32+c] * Q[node][h*32+c] ----------------
__global__ __launch_bounds__(128)
void k_gproj(const float* __restrict__ Q, const float* __restrict__ We,
             float* __restrict__ G, int N)
{
    __shared__ float WeS[ED_DIM * D_DIM];
    for (int i = threadIdx.x; i < ED_DIM * D_DIM; i += 128) WeS[i] = We[i];
    __syncthreads();
    int node = blockIdx.x * 128 + threadIdx.x;
    if (node >= N) return;
    float q[D_DIM];
    const float4* qp = (const float4*)(Q + (size_t)node * D_DIM);
    #pragma unroll
    for (int i = 0; i < D_DIM / 4; ++i) {
        float4 v = qp[i];
        q[4*i] = v.x; q[4*i+1] = v.y; q[4*i+2] = v.z; q[4*i+3] = v.w;
    }
    #pragma unroll
    for (int h = 0; h < H_HEADS; ++h) {
        #pragma unroll
        for (int j = 0; j < ED_DIM; ++j) {
            float s = 0.f;
            #pragma unroll
            for (int c = 0; c < C_CH; ++c) s += q[h*C_CH + c] * WeS[j*D_DIM + h*C_CH + c];
            G[(size_t)node * (H_HEADS*ED_DIM) + h*ED_DIM + j] = s;
        }
    }
}

// ---------------- K3: per-edge alpha + deterministic segment max ----------------
__global__ __launch_bounds__(256)
void k_alpha(const long long* __restrict__ ei, const float* __restrict__ et,
             const float* __restrict__ msg, const long long* __restrict__ lu,
             const float* __restrict__ tw, const float* __restrict__ tb,
             const float* __restrict__ Q, const float* __restrict__ K,
             const float* __restrict__ G,
             float* __restrict__ alphaBuf, unsigned* __restrict__ amaxU, int E)
{
    int e = blockIdx.x * 256 + threadIdx.x;
    if (e >= E) return;
    int src = (int)ei[e];
    int dst = (int)ei[(size_t)E + e];

    const float* qd = Q + (size_t)dst * D_DIM;
    const float* ks = K + (size_t)src * D_DIM;
    const float* gd = G + (size_t)dst * (H_HEADS*ED_DIM);
    // kick the L2-resident gathers while we burn VALU on the cosine chain
    __builtin_prefetch(qd, 0, 0);
    __builtin_prefetch((const char*)qd + 128, 0, 0);
    __builtin_prefetch(ks, 0, 0);
    __builtin_prefetch((const char*)ks + 128, 0, 0);
    __builtin_prefetch(gd, 0, 0);

    float rel = fmaxf(et[e] - (float)lu[src], 0.0f);
    float attr[ED_DIM];
    #pragma unroll
    for (int j = 0; j < TD_DIM; ++j) attr[j] = __cosf(rel * tw[j] + tb[j]);
    attr[TD_DIM] = msg[e];

    #pragma unroll
    for (int h = 0; h < H_HEADS; ++h) {
        float s = 0.f;
        #pragma unroll
        for (int c = 0; c < C_CH; c += 4) {
            float4 qv = *(const float4*)(qd + h*C_CH + c);
            float4 kv = *(const float4*)(ks + h*C_CH + c);
            s += qv.x*kv.x + qv.y*kv.y + qv.z*kv.z + qv.w*kv.w;
        }
        #pragma unroll
        for (int j = 0; j < ED_DIM; ++j) s += attr[j] * gd[h*ED_DIM + j];
        float a = s * 0.17677669529663689f;   // 1/sqrt(32)
        alphaBuf[(size_t)e * H_HEADS + h] = a;
        atomicMax(&amaxU[dst * H_HEADS + h], fkey(a));
    }
}

// ---------------- K4: ex = exp(alpha - amax[dst]); denom += ex (fixed point) ----------------
__global__ __launch_bounds__(256)
void k_exps(const long long* __restrict__ ei, const unsigned* __restrict__ amaxU,
            float* __restrict__ exBuf, unsigned long long* __restrict__ denomFix, int E)
{
    int e = blockIdx.x * 256 + threadIdx.x;
    if (e >= E) return;
    int dst = (int)ei[(size_t)E + e];
    #pragma unroll
    for (int h = 0; h < H_HEADS; ++h) {
        float m  = funkey(amaxU[dst * H_HEADS + h]);
        float a  = exBuf[(size_t)e * H_HEADS + h];
        float ex = __expf(a - m);
        exBuf[(size_t)e * H_HEADS + h] = ex;
        atomicAdd(&denomFix[dst * H_HEADS + h], f2fix(ex));
    }
}

// ---------------- K5: scatter attn*v[src] and attn*attr_j (fixed point) ----------------
__global__ __launch_bounds__(256)
void k_scatter(const long long* __restrict__ ei, const float* __restrict__ et,
               const float* __restrict__ msg, const long long* __restrict__ lu,
               const float* __restrict__ tw, const float* __restrict__ tb,
               const float* __restrict__ V, const float* __restrict__ exBuf,
               const unsigned long long* __restrict__ denomFix,
               unsigned long long* __restrict__ SAcc,
               unsigned long long* __restrict__ OutAcc, int E)
{
    int e = blockIdx.x * 256 + threadIdx.x;
    if (e >= E) return;
    int src = (int)ei[e];
    int dst = (int)ei[(size_t)E + e];

    const float* vs = V + (size_t)src * D_DIM;
    __builtin_prefetch(vs, 0, 0);
    __builtin_prefetch((const char*)vs + 128, 0, 0);

    float rel = fmaxf(et[e] - (float)lu[src], 0.0f);
    float attr[ED_DIM];
    #pragma unroll
    for (int j = 0; j < TD_DIM; ++j) attr[j] = __cosf(rel * tw[j] + tb[j]);
    attr[TD_DIM] = msg[e];

    #pragma unroll
    for (int h = 0; h < H_HEADS; ++h) {
        float den  = fix2f(denomFix[dst * H_HEADS + h]) + 1e-16f;
        float attn = exBuf[(size_t)e * H_HEADS + h] / den;
        #pragma unroll
        for (int j = 0; j < ED_DIM; ++j)
            atomicAdd(&SAcc[(size_t)dst * (H_HEADS*ED_DIM) + h*ED_DIM + j],
                      f2fix(attn * attr[j]));
        #pragma unroll
        for (int c = 0; c < C_CH; c += 4) {
            float4 vv = *(const float4*)(vs + h*C_CH + c);
            size_t ob = (size_t)dst * D_DIM + h*C_CH + c;
            atomicAdd(&OutAcc[ob + 0], f2fix(attn * vv.x));
            atomicAdd(&OutAcc[ob + 1], f2fix(attn * vv.y));
            atomicAdd(&OutAcc[ob + 2], f2fix(attn * vv.z));
            atomicAdd(&OutAcc[ob + 3], f2fix(attn * vv.w));
        }
    }
}

// ---------------- K6: out = skip (already in d_out) + fix(OutAcc) + S @ We ----------------
__global__ __launch_bounds__(256)
void k_final(const unsigned long long* __restrict__ OutAcc,
             const unsigned long long* __restrict__ SAcc,
             const float* __restrict__ We, float* __restrict__ out, int N)
{
    int idx = blockIdx.x * 256 + threadIdx.x;
    if (idx >= N * D_DIM) return;
    int node = idx >> 6;
    int n    = idx & 63;
    int h    = n >> 5;
    const unsigned long long* sp = SAcc + (size_t)node * (H_HEADS*ED_DIM) + h*ED_DIM;
    float se = 0.f;
    #pragma unroll
    for (int j = 0; j < ED_DIM; ++j) se += fix2f(sp[j]) * We[j*D_DIM + n];
    out[idx] += fix2f(OutAcc[idx]) + se;
}

extern "C" void kernel_launch(void* const* d_in, const int* in_sizes, int n_in,
                              void* d_out, int out_size, void* d_ws, size_t ws_size,
                              hipStream_t stream) {
    const float*      X   = (const float*)d_in[0];        // node_memory [N,64]
    const long long*  lu  = (const long long*)d_in[1];    // last_update [N]
    const long long*  ei  = (const long long*)d_in[2];    // edge_index [2,E]
    const float*      et  = (const float*)d_in[3];        // event_t [E]
    const float*      msg = (const float*)d_in[4];        // event_msg [E,1]
    const float*      tw  = (const float*)d_in[5];
    const float*      tb  = (const float*)d_in[6];
    const float*      Wq  = (const float*)d_in[7];
    const float*      bq  = (const float*)d_in[8];
    const float*      Wk  = (const float*)d_in[9];
    const float*      bk  = (const float*)d_in[10];
    const float*      Wv  = (const float*)d_in[11];
    const float*      bv  = (const float*)d_in[12];
    const float*      We  = (const float*)d_in[13];       // [17,64]
    const float*      Ws  = (const float*)d_in[14];
    const float*      bs  = (const float*)d_in[15];
    float* out = (float*)d_out;

    const int N = in_sizes[0] / D_DIM;
    const int E = in_sizes[3];

    // ---- workspace carve-up (256B aligned) ----
    char* ws = (char*)d_ws;
    size_t off = 0;
    auto carve = [&](size_t bytes) -> char* {
        char* p = ws + off;
        off = (off + bytes + 255) & ~(size_t)255;
        return p;
    };
    float* Q  = (float*)carve((size_t)N * D_DIM * 4);
    float* K  = (float*)carve((size_t)N * D_DIM * 4);
    float* V  = (float*)carve((size_t)N * D_DIM * 4);
    float* G  = (float*)carve((size_t)N * H_HEADS * ED_DIM * 4);
    float* exBuf = (float*)carve((size_t)E * H_HEADS * 4);   // alpha, then ex
    size_t zbase = off;                                      // zero-initialized region:
    unsigned*           amaxU    = (unsigned*)carve((size_t)N * H_HEADS * 4);
    unsigned long long* denomFix = (unsigned long long*)carve((size_t)N * H_HEADS * 8);
    unsigned long long* SAcc     = (unsigned long long*)carve((size_t)N * H_HEADS * ED_DIM * 8);
    unsigned long long* OutAcc   = (unsigned long long*)carve((size_t)N * D_DIM * 8);
    size_t zbytes = off - zbase;

    // K0: zero the atomic accumulators
    size_t n64 = zbytes / 8;
    int zblocks = (int)((n64 + 255) / 256);
    if (zblocks > 16384) zblocks = 16384;
    k_zero<<<zblocks, 256, 0, stream>>>((unsigned long long*)(ws + zbase), n64);

    // K1: Q,K,V + skip(->d_out) via WMMA (async LDS staging)
    k_qkvs<<<(N + 31) / 32, 256, 0, stream>>>(X, Wq, bq, Wk, bk, Wv, bv, Ws, bs,
                                              Q, K, V, out, N);
    // K2: G projection
    k_gproj<<<(N + 127) / 128, 128, 0, stream>>>(Q, We, G, N);
    // K3: alpha + segment max
    k_alpha<<<(E + 255) / 256, 256, 0, stream>>>(ei, et, msg, lu, tw, tb, Q, K, G,
                                                 exBuf, amaxU, E);
    // K4: exp + denom
    k_exps<<<(E + 255) / 256, 256, 0, stream>>>(ei, amaxU, exBuf, denomFix, E);
    // K5: scatter
    k_scatter<<<(E + 255) / 256, 256, 0, stream>>>(ei, et, msg, lu, tw, tb, V, exBuf,
                                                   denomFix, SAcc, OutAcc, E);
    // K6: finalize
    k_final<<<((N * D_DIM) + 255) / 256, 256, 0, stream>>>(OutAcc, SAcc, We, out, N);
}